// ULayer_18726057410941
// MI455X (gfx1250) — compile-verified
//
#include <hip/hip_runtime.h>

typedef __attribute__((ext_vector_type(2))) float v2f;
typedef __attribute__((ext_vector_type(8))) float v8f;

#define DIM 4096   // 2^12
#define NTHREADS 256

// ---------------------------------------------------------------------------
// Kernel 1: build the three 16x16 diamond-gate matrices (re/im planes) from
// thetas into workspace.  Layout: uws[g*512 + r*16 + c] = Re(U_g[r][c]),
//                                 uws[g*512 + 256 + r*16 + c] = Im(U_g[r][c]).
// ---------------------------------------------------------------------------
__global__ __launch_bounds__(256) void build_gates_kernel(
    const float* __restrict__ thetas, float* __restrict__ uws) {
  const int g = blockIdx.x;           // gate 0..2
  const int t = threadIdx.x;          // 0..255 -> (row, col)
  const int r = t >> 4;
  const int c = t & 15;

  const float th = thetas[g];
  const float cs = cosf(th);
  const float sn = sinf(th);

  float re = 0.0f, im = 0.0f;
  if ((r >> 2) == (c >> 2)) {         // block-diagonal over control sector
    const int k = r >> 2;             // control sector 0..3
    const int i = r & 3, j = c & 3;   // local target indices
    if (k == 0 || k == 3) {
      // XY rotation: [[1,0,0,0],[0,c,s,0],[0,s,c,0],[0,0,0,e^{+-i th}]]
      // k==0: s = -i sin, phase e^{-i th}; k==3: s = +i sin, phase e^{+i th}
      const float sgn = (k == 0) ? -1.0f : 1.0f;
      if (i == 0 && j == 0) { re = 1.0f; }
      else if (i == 3 && j == 3) { re = cs; im = sgn * sn; }
      else if (i == j) { re = cs; }                              // (1,1),(2,2)
      else if ((i == 1 && j == 2) || (i == 2 && j == 1)) { im = sgn * sn; }
    } else {
      // diag(1, e^{+-i th}, e^{+-i th}, 1); k==1 -> e^{-i th}, k==2 -> e^{+i th}
      const float sgn = (k == 1) ? -1.0f : 1.0f;
      if (i == j) {
        if (i == 0 || i == 3) re = 1.0f;
        else { re = cs; im = sgn * sn; }
      }
    }
  }
  uws[g * 512 + r * 16 + c]       = re;
  uws[g * 512 + 256 + r * 16 + c] = im;
}

// ---------------------------------------------------------------------------
// Kernel 2: one workgroup per state vector.  Vector (re+im) lives in LDS.
// Apply U0 (axis a), U1 (axis b), U2 (axis c) as 16 complex 16x16 GEMM tiles
// each, built from V_WMMA_F32_16X16X4_F32 (complex = 4 real GEMMs; K=16 done
// as 4 chained K=4 WMMAs).  In-place per tile; barriers between passes.
// ---------------------------------------------------------------------------
__global__ __launch_bounds__(NTHREADS) void apply_ulayer_kernel(
    const float* __restrict__ in_re, const float* __restrict__ in_im,
    const float* __restrict__ uws, float* __restrict__ out) {
  __shared__ float sre[DIM];
  __shared__ float sim[DIM];

  const int tid = threadIdx.x;
  const int b   = blockIdx.x;

  // Stage the whole 4096-complex vector into LDS (coalesced).
  for (int i = tid; i < DIM; i += NTHREADS) {
    sre[i] = in_re[b * DIM + i];
    sim[i] = in_im[b * DIM + i];
  }
  __syncthreads();

  const int lane = tid & 31;          // wave32
  const int wave = tid >> 5;          // 0..7
  const int half = lane >> 4;         // 0 | 1
  const int lrow = lane & 15;         // A: row m / B: col n / D: col n
  const int kcb  = half * 2;          // A/B: local-k base per lane half

  // Per-pass strides for addr = tile*tS + k*kS + n*nS  (k also = output row m)
  // pass0: contract a (s[a,b,c] = a*256+b*16+c), tiles over bc-hi
  // pass1: contract b, tiles over a
  // pass2: contract c, tiles over a (columns = b)
  const int TsA[3] = {16, 256, 256};
  const int KsA[3] = {256, 16, 1};
  const int NsA[3] = {1, 1, 16};

  for (int g = 0; g < 3; ++g) {
    // ---- A fragments: gate g, 16x16 split into four 16x4 K-slices ----
    const float* __restrict__ ur = uws + g * 512;
    const float* __restrict__ ui = ur + 256;
    v2f aR[4], aI[4], aIn[4];
#pragma unroll
    for (int kk = 0; kk < 4; ++kk) {
      const int k0 = kk * 4 + kcb;
      v2f r, m;
      r[0] = ur[lrow * 16 + k0];
      r[1] = ur[lrow * 16 + k0 + 1];
      m[0] = ui[lrow * 16 + k0];
      m[1] = ui[lrow * 16 + k0 + 1];
      aR[kk] = r;
      aI[kk] = m;
      aIn[kk] = -m;                   // pre-negated Ai (f32 WMMA has no A-neg)
    }

    const int tS = TsA[g], kS = KsA[g], nS = NsA[g];

#pragma unroll
    for (int tt = 0; tt < 2; ++tt) {
      const int tile = wave + tt * 8; // 8 waves x 2 = 16 tiles
      const int base = tile * tS + lrow * nS;

      // ---- B fragments: the wave's own 16x16 input tile from LDS ----
      v2f bR[4], bI[4];
#pragma unroll
      for (int kk = 0; kk < 4; ++kk) {
        const int k0 = kk * 4 + kcb;
        v2f r, m;
        r[0] = sre[base + k0 * kS];
        r[1] = sre[base + (k0 + 1) * kS];
        m[0] = sim[base + k0 * kS];
        m[1] = sim[base + (k0 + 1) * kS];
        bR[kk] = r;
        bI[kk] = m;
      }

      // ---- Complex GEMM: 16 x v_wmma_f32_16x16x4_f32 ----
      v8f cR = {};
      v8f cI = {};
#pragma unroll
      for (int kk = 0; kk < 4; ++kk) {
        cR = __builtin_amdgcn_wmma_f32_16x16x4_f32(
            false, aR[kk],  false, bR[kk], (short)0, cR, false, false);
        cR = __builtin_amdgcn_wmma_f32_16x16x4_f32(
            false, aIn[kk], false, bI[kk], (short)0, cR, false, false);
        cI = __builtin_amdgcn_wmma_f32_16x16x4_f32(
            false, aR[kk],  false, bI[kk], (short)0, cI, false, false);
        cI = __builtin_amdgcn_wmma_f32_16x16x4_f32(
            false, aI[kk],  false, bR[kk], (short)0, cI, false, false);
      }

      // ---- Write D back in place (tile output aliases tile input) ----
#pragma unroll
      for (int j = 0; j < 8; ++j) {
        const int m = j + half * 8;             // D row per VGPR layout
        const int addr = tile * tS + m * kS + lrow * nS;
        sre[addr] = cR[j];
        sim[addr] = cI[j];
      }
    }
    __syncthreads();
  }

  // Emit [B, 4096, 2] = interleaved (re, im) -> one float2 store per element.
  float2* __restrict__ out2 = (float2*)out;
  for (int i = tid; i < DIM; i += NTHREADS) {
    out2[b * DIM + i] = make_float2(sre[i], sim[i]);
  }
}

// ---------------------------------------------------------------------------
extern "C" void kernel_launch(void* const* d_in, const int* in_sizes, int n_in,
                              void* d_out, int out_size, void* d_ws, size_t ws_size,
                              hipStream_t stream) {
  const float* in_re  = (const float*)d_in[0];
  const float* in_im  = (const float*)d_in[1];
  const float* thetas = (const float*)d_in[2];
  float* uws = (float*)d_ws;                     // 3 * 512 floats = 6 KB
  float* out = (float*)d_out;

  const int batch = in_sizes[0] / DIM;           // 256

  build_gates_kernel<<<3, 256, 0, stream>>>(thetas, uws);
  apply_ulayer_kernel<<<batch, NTHREADS, 0, stream>>>(in_re, in_im, uws, out);
}